// GRU_89567247991392
// MI455X (gfx1250) — compile-verified
//
#include <hip/hip_runtime.h>
#include <hip/hip_bf16.h>
#include <stdint.h>
#include <stddef.h>

// Problem constants from the reference: (T, B, I, H, O) = (4096, 8192, 2, 1, 1)
#define T_LEN 4096
#define B_SZ  8192
#define CHUNK 16   // timesteps prefetched per double-buffer stage (T % CHUNK == 0)
#define BLK   32   // one wave per workgroup -> spread across max WGPs/SIMDs

typedef float f32x2 __attribute__((ext_vector_type(2)));
typedef int   v2i   __attribute__((vector_size(8)));   // matches builtin's V2i

// ---------------- gfx1250 hardware tanh (TRANS op) ----------------
__device__ __forceinline__ float tanh_hw(float x) {
#if defined(__HIP_DEVICE_COMPILE__) && __has_builtin(__builtin_amdgcn_tanhf)
  return __builtin_amdgcn_tanhf(x);
#elif defined(__HIP_DEVICE_COMPILE__)
  float r;
  asm("v_tanh_f32 %0, %1" : "=v"(r) : "v"(x));
  return r;
#else
  return tanhf(x);  // host parse only; never executed
#endif
}

// ---------------- CDNA5 async global->LDS copy path ----------------
#if defined(__HIP_DEVICE_COMPILE__) && defined(__gfx1250__) && \
    __has_builtin(__builtin_amdgcn_global_load_async_to_lds_b64)
#define USE_ASYNC 1
#else
#define USE_ASYNC 0
#endif

#if USE_ASYNC
// Builtin signature (from clang diagnostic): (v2i AS1* src, v2i AS3* dst, Ii, Ii)
__device__ __forceinline__ void async_copy_b64(const void* gsrc, void* ldst) {
  __builtin_amdgcn_global_load_async_to_lds_b64(
      (__attribute__((address_space(1))) v2i*)(gsrc),
      (__attribute__((address_space(3))) v2i*)(ldst),
      0, 0);
}
template <int N>
__device__ __forceinline__ void wait_async() {
#if __has_builtin(__builtin_amdgcn_s_wait_asynccnt)
  __builtin_amdgcn_s_wait_asynccnt(N);
#else
  asm volatile("s_wait_asynccnt %0" ::"i"(N) : "memory");
#endif
}
#endif

// One thread owns one batch element's full T-step recurrence.
// sigmoid(x) = 0.5*tanh(x/2)+0.5 folded into pre-halved weights:
//   critical path per step = fma -> v_tanh -> fma -> v_tanh -> 2 fma.
__global__ void __launch_bounds__(BLK) gru_scan_kernel(
    const float* __restrict__ x,     // (T, B, 2)
    const float* __restrict__ hinit, // (1, B, 1)
    const float* __restrict__ Wih,   // (3, 2) rows: r, z, n
    const float* __restrict__ Whh,   // (3, 1)
    const float* __restrict__ Wout,  // (1, 1)
    const float* __restrict__ bo,    // (1,)
    float* __restrict__ out)         // (T*B) output, then (B) last_h
{
  const int tid = threadIdx.x;
  const int b   = blockIdx.x * BLK + tid;

  // Uniform scalar weights (loaded once; pre-scale by 0.5 for sigmoid-as-tanh).
  const float a00h = 0.5f * Wih[0], a01h = 0.5f * Wih[1];   // r-gate input weights /2
  const float a10h = 0.5f * Wih[2], a11h = 0.5f * Wih[3];   // z-gate input weights /2
  const float a20  = Wih[4],        a21  = Wih[5];          // n-gate input weights
  const float wr2 = 0.5f * Whh[0], wz2 = 0.5f * Whh[1], wn2 = 0.5f * Whh[2];
  const float wo = Wout[0], bout = bo[0];

  const f32x2* __restrict__ xin = (const f32x2*)x;  // element index = t*B + b
  float h = hinit[b];

#if USE_ASYNC
  __shared__ f32x2 lbuf[2][CHUNK][BLK];  // 8 KB double buffer, ASYNCcnt-tracked
  #pragma unroll
  for (int j = 0; j < CHUNK; ++j)
    async_copy_b64(&xin[(size_t)j * B_SZ + b], &lbuf[0][j][tid]);
#else
  f32x2 cur[CHUNK];
  #pragma unroll
  for (int j = 0; j < CHUNK; ++j)
    cur[j] = __builtin_nontemporal_load(&xin[(size_t)j * B_SZ + b]);
#endif

  for (int tc = 0; tc < T_LEN; tc += CHUNK) {
#if USE_ASYNC
    const int p = (tc / CHUNK) & 1;
    if (tc + CHUNK < T_LEN) {
      #pragma unroll
      for (int j = 0; j < CHUNK; ++j)
        async_copy_b64(&xin[(size_t)(tc + CHUNK + j) * B_SZ + b],
                       &lbuf[p ^ 1][j][tid]);
      wait_async<CHUNK>();  // previous CHUNK (current buffer) complete, next in flight
    } else {
      wait_async<0>();
    }
    asm volatile("" ::: "memory");  // don't let LDS reads float above the wait
#else
    f32x2 nxt[CHUNK];
    if (tc + CHUNK < T_LEN) {
      #pragma unroll
      for (int j = 0; j < CHUNK; ++j)
        nxt[j] = __builtin_nontemporal_load(&xin[(size_t)(tc + CHUNK + j) * B_SZ + b]);
    }
#endif

    #pragma unroll
    for (int j = 0; j < CHUNK; ++j) {
#if USE_ASYNC
      const f32x2 v = lbuf[p][j][tid];   // each lane reads only its own slot
#else
      const f32x2 v = cur[j];
#endif
      const float x0 = v.x, x1 = v.y;
      // x-gate contributions (off the h critical path)
      const float xr2 = fmaf(x1, a01h, x0 * a00h);       // (xr)/2
      const float xz2 = fmaf(x1, a11h, x0 * a10h);       // (xz)/2
      const float xn  = fmaf(x1, a21,  x0 * a20);
      // r = sigmoid(xr + h*wr) = 0.5*tr + 0.5 ; tr = tanh((xr + h*wr)/2)
      const float tr  = tanh_hw(fmaf(h, wr2, xr2));
      const float tz  = tanh_hw(fmaf(h, wz2, xz2));      // z chain, parallel to r
      const float hn2 = h * wn2;                         // 0.5 * h * wn
      // n = tanh(xn + r*h*wn) = tanh((xn + hn2) + tr*hn2)
      const float tn  = tanh_hw(fmaf(tr, hn2, xn + hn2));
      // h' = (1-z)*n + z*h = 0.5*(h+n) + tz*0.5*(h-n)
      const float hh  = 0.5f * h;
      const float hp  = fmaf(tn,  0.5f, hh);
      const float hm  = fmaf(tn, -0.5f, hh);
      h = fmaf(tz, hm, hp);
      __builtin_nontemporal_store(fmaf(h, wo, bout),
                                  &out[(size_t)(tc + j) * B_SZ + b]);
    }

#if !USE_ASYNC
    if (tc + CHUNK < T_LEN) {
      #pragma unroll
      for (int j = 0; j < CHUNK; ++j) cur[j] = nxt[j];
    }
#endif
  }

  out[(size_t)T_LEN * B_SZ + b] = h;  // last_h appended after (T*B) outputs
}

extern "C" void kernel_launch(void* const* d_in, const int* in_sizes, int n_in,
                              void* d_out, int out_size, void* d_ws, size_t ws_size,
                              hipStream_t stream) {
  (void)in_sizes; (void)n_in; (void)out_size; (void)d_ws; (void)ws_size;
  const float* input = (const float*)d_in[0];
  const float* h0    = (const float*)d_in[1];
  const float* Wih   = (const float*)d_in[2];
  const float* Whh   = (const float*)d_in[3];
  const float* Wout  = (const float*)d_in[4];
  const float* bout  = (const float*)d_in[5];
  float* out = (float*)d_out;

  dim3 grid(B_SZ / BLK);   // 256 single-wave workgroups -> max WGP spread
  dim3 block(BLK);
  hipLaunchKernelGGL(gru_scan_kernel, grid, block, 0, stream,
                     input, h0, Wih, Whh, Wout, bout, out);
}